// DirectImageAlign_42863773614750
// MI455X (gfx1250) — compile-verified
//
#include <hip/hip_runtime.h>
#include <math.h>

typedef __attribute__((ext_vector_type(2))) float v2f;
typedef __attribute__((ext_vector_type(8))) float v8f;

#define B_ 4
#define C_ 8
#define H_ 192
#define W_ 256
#define N_ (H_*W_)
#define GEOMW 0.01f
#define HUBER_D 0.5f
#define LM_EPS 1e-6f

// ---------------------------------------------------------------------------
// Precompute: vertex maps V0/V1, normal maps N0/N1, image gradients GX/GY
// ---------------------------------------------------------------------------
__device__ inline void vertex_at(int x, int y, const float* dptr,
                                 float fx, float fy, float cx, float cy,
                                 float out[3]) {
  float d = dptr[y*W_ + x];
  out[0] = ((float)x - cx) / fx * d;
  out[1] = ((float)y - cy) / fy * d;
  out[2] = d;
}

__global__ __launch_bounds__(256) void precompute_kernel(
    const float* __restrict__ I0, const float* __restrict__ depth0,
    const float* __restrict__ depth1, const float* __restrict__ intr,
    float* __restrict__ V0, float* __restrict__ V1,
    float* __restrict__ N0, float* __restrict__ N1,
    float* __restrict__ GX, float* __restrict__ GY) {
  int gid = blockIdx.x * blockDim.x + threadIdx.x;
  if (gid >= B_ * N_) return;
  int b = gid / N_, p = gid - b * N_;
  int y = p / W_, x = p - y * W_;
  float fx = intr[b*4+0], fy = intr[b*4+1], cx = intr[b*4+2], cy = intr[b*4+3];
  int xm = max(x-1, 0), xp = min(x+1, W_-1);
  int ym = max(y-1, 0), yp = min(y+1, H_-1);

  const float* dmaps[2] = { depth0 + b*N_, depth1 + b*N_ };
  float* Vout[2] = { V0 + b*3*N_, V1 + b*3*N_ };
  float* Nout[2] = { N0 + b*3*N_, N1 + b*3*N_ };
  #pragma unroll
  for (int s = 0; s < 2; ++s) {
    float vc[3], va[3], vb[3], dxv[3], dyv[3];
    vertex_at(x, y, dmaps[s], fx, fy, cx, cy, vc);
    Vout[s][0*N_+p] = vc[0]; Vout[s][1*N_+p] = vc[1]; Vout[s][2*N_+p] = vc[2];
    vertex_at(xp, y, dmaps[s], fx, fy, cx, cy, va);
    vertex_at(xm, y, dmaps[s], fx, fy, cx, cy, vb);
    dxv[0] = (va[0]-vb[0])*0.5f; dxv[1] = (va[1]-vb[1])*0.5f; dxv[2] = (va[2]-vb[2])*0.5f;
    vertex_at(x, yp, dmaps[s], fx, fy, cx, cy, va);
    vertex_at(x, ym, dmaps[s], fx, fy, cx, cy, vb);
    dyv[0] = (va[0]-vb[0])*0.5f; dyv[1] = (va[1]-vb[1])*0.5f; dyv[2] = (va[2]-vb[2])*0.5f;
    float nx = dxv[1]*dyv[2] - dxv[2]*dyv[1];
    float ny = dxv[2]*dyv[0] - dxv[0]*dyv[2];
    float nz = dxv[0]*dyv[1] - dxv[1]*dyv[0];
    float inv = 1.f / sqrtf(nx*nx + ny*ny + nz*nz + 1e-12f);
    Nout[s][0*N_+p] = nx*inv; Nout[s][1*N_+p] = ny*inv; Nout[s][2*N_+p] = nz*inv;
  }

  const float* I0b = I0 + b*C_*N_;
  float* GXb = GX + b*C_*N_;
  float* GYb = GY + b*C_*N_;
  #pragma unroll
  for (int c = 0; c < C_; ++c) {
    const float* F = I0b + c*N_;
    GXb[c*N_+p] = (F[y*W_+xp] - F[y*W_+xm]) * 0.5f;
    GYb[c*N_+p] = (F[yp*W_+x] - F[ym*W_+x]) * 0.5f;
  }
}

// ---------------------------------------------------------------------------
// Bilinear sampling with per-corner out-of-bounds zeroing (matches reference)
// ---------------------------------------------------------------------------
struct BL4 { int i0, i1, i2, i3; float w0, w1, w2, w3; };

__device__ inline BL4 mk_bl(float u, float v) {
  BL4 bl;
  float u0 = floorf(u), v0f = floorf(v);
  float du = u - u0, dv = v - v0f;
  float uu[2] = { u0, u0 + 1.f }, vv[2] = { v0f, v0f + 1.f };
  float wu[2] = { 1.f - du, du }, wv[2] = { 1.f - dv, dv };
  int idxs[4]; float ws[4];
  #pragma unroll
  for (int j = 0; j < 2; ++j) {
    #pragma unroll
    for (int i = 0; i < 2; ++i) {
      float uf = uu[i], vf = vv[j];
      float inb = (uf >= 0.f && uf <= (float)(W_-1) &&
                   vf >= 0.f && vf <= (float)(H_-1)) ? 1.f : 0.f;
      int ui = (int)fminf(fmaxf(uf, 0.f), (float)(W_-1));
      int vi = (int)fminf(fmaxf(vf, 0.f), (float)(H_-1));
      idxs[j*2+i] = vi * W_ + ui;
      ws[j*2+i] = wu[i] * wv[j] * inb;
    }
  }
  bl.i0 = idxs[0]; bl.i1 = idxs[1]; bl.i2 = idxs[2]; bl.i3 = idxs[3];
  bl.w0 = ws[0];   bl.w1 = ws[1];   bl.w2 = ws[2];   bl.w3 = ws[3];
  return bl;
}

__device__ inline float bl_sample(const float* __restrict__ F, const BL4& bl) {
  return F[bl.i0]*bl.w0 + F[bl.i1]*bl.w1 + F[bl.i2]*bl.w2 + F[bl.i3]*bl.w3;
}

// ---------------------------------------------------------------------------
// Fused residual/Jacobian + WMMA Gram-matrix (JtWJ | JtWr) accumulation.
// Block = 128 threads = 4 waves; 1 pixel per lane; each wave reduces its 32
// pixels' 288 augmented rows with 72 x V_WMMA_F32_16X16X4_F32 (A==B fragment,
// symmetric product). Rows are 8 floats: [J0..J5, r, 0]; the zero pad slot is
// read by lanes with idx>=7 (clamped column) so every ds_load is unconditional
// and EXEC stays all-ones through the WMMA loop (no saveexec churn).
// ---------------------------------------------------------------------------
__global__ __launch_bounds__(128) void accum_kernel(
    const float* __restrict__ I0, const float* __restrict__ I1,
    const float* __restrict__ depth0, const float* __restrict__ depth1,
    const float* __restrict__ intr, const float* __restrict__ poseB,
    const float* __restrict__ V0, const float* __restrict__ V1,
    const float* __restrict__ N0, const float* __restrict__ N1,
    const float* __restrict__ GX, const float* __restrict__ GY,
    float* __restrict__ gacc) {
  __shared__ float lds[4 * 288 * 8];
  __shared__ float bacc[49];
  const int tid = threadIdx.x;
  const int b = blockIdx.y;
  const int wave = tid >> 5, lane = tid & 31;
  const int p = blockIdx.x * 128 + tid;

  if (tid < 49) bacc[tid] = 0.f;

  // current pose
  const float* P = poseB + b * 16;
  float R00 = P[0],  R01 = P[1],  R02 = P[2],  t0 = P[3];
  float R10 = P[4],  R11 = P[5],  R12 = P[6],  t1 = P[7];
  float R20 = P[8],  R21 = P[9],  R22 = P[10], t2 = P[11];
  float fx = intr[b*4+0], fy = intr[b*4+1], cx = intr[b*4+2], cy = intr[b*4+3];

  const float* V0b = V0 + b*3*N_;
  const float* V1b = V1 + b*3*N_;
  const float* N0b = N0 + b*3*N_;

  float vx = V1b[0*N_+p], vy = V1b[1*N_+p], vz = V1b[2*N_+p];
  float Xx = R00*vx + R01*vy + R02*vz + t0;
  float Xy = R10*vx + R11*vy + R12*vz + t1;
  float Xz = R20*vx + R21*vy + R22*vz + t2;
  float zs  = (fabsf(Xz) > 1e-8f) ? Xz : 1e-8f;
  float izs = 1.f / zs;
  float u = fx * Xx * izs + cx;
  float v = fy * Xy * izs + cy;
  bool inb    = (u > 0.f) && (u < (float)(W_-1)) && (v > 0.f) && (v < (float)(H_-1));
  bool valid0 = Xz > 1e-8f;
  bool vd     = (depth0[b*N_+p] > 0.f) && (depth1[b*N_+p] > 0.f);
  float vmask = (inb && valid0 && vd) ? 1.f : 0.f;

  // Jw = Jp (2x3 projection) * [-skew(X) | I3] (3x6)
  float Jp00 = fx*izs, Jp02 = -fx*Xx*izs*izs;
  float Jp11 = fy*izs, Jp12 = -fy*Xy*izs*izs;
  float Jw0[6] = {  Jp02*Xy, Jp00*Xz - Jp02*Xx, -Jp00*Xy, Jp00, 0.f,  Jp02 };
  float Jw1[6] = { -Jp11*Xz + Jp12*Xy, -Jp12*Xx,  Jp11*Xx, 0.f,  Jp11, Jp12 };

  // photometric grid coords (the +-2 grid clip == clamp in pixel space)
  float uq = fminf(fmaxf(u, -0.5f*(float)(W_-1)), 1.5f*(float)(W_-1));
  float vq = fminf(fmaxf(v, -0.5f*(float)(H_-1)), 1.5f*(float)(H_-1));
  BL4 blq = mk_bl(uq, vq);   // photometric (clipped)
  BL4 blr = mk_bl(u, v);     // ICP (raw)

  float* myrows = lds + (wave*288 + lane*9) * 8;
  const float* I0b = I0 + b*C_*N_;
  const float* I1b = I1 + b*C_*N_;
  const float* GXb = GX + b*C_*N_;
  const float* GYb = GY + b*C_*N_;

  #pragma unroll
  for (int c = 0; c < C_; ++c) {
    float I0w = bl_sample(I0b + c*N_, blq);
    float gx  = bl_sample(GXb + c*N_, blq);
    float gy  = bl_sample(GYb + c*N_, blq);
    float rr  = (I1b[c*N_+p] - I0w) * vmask;
    float ar  = fabsf(rr);
    float hw  = (ar <= HUBER_D) ? 1.f : HUBER_D / fmaxf(ar, 1e-12f);
    float s   = sqrtf(hw);
    #pragma unroll
    for (int k = 0; k < 6; ++k)
      myrows[c*8 + k] = s * (gx*Jw0[k] + gy*Jw1[k]) * vmask;
    myrows[c*8 + 6] = s * rr;
    myrows[c*8 + 7] = 0.f;          // zero pad read by lanes idx>=7
  }

  // ICP (point-to-plane) row
  {
    float rVx = bl_sample(V0b + 0*N_, blr);
    float rVy = bl_sample(V0b + 1*N_, blr);
    float rVz = bl_sample(V0b + 2*N_, blr);
    float nx  = bl_sample(N0b + 0*N_, blr);
    float ny  = bl_sample(N0b + 1*N_, blr);
    float nz  = bl_sample(N0b + 2*N_, blr);
    float dx = Xx - rVx, dy = Xy - rVy, dz = Xz - rVz;
    float dn = sqrtf(dx*dx + dy*dy + dz*dz + 1e-12f);
    bool occ = (!inb) || (dn > 0.1f);
    float res = nx*dx + ny*dy + nz*dz;
    float NtC0 = nx*R00 + ny*R10 + nz*R20;
    float NtC1 = nx*R01 + ny*R11 + nz*R21;
    float NtC2 = nx*R02 + ny*R12 + nz*R22;
    float Jr0 =  NtC1*vz - NtC2*vy;
    float Jr1 = -NtC0*vz + NtC2*vx;
    float Jr2 =  NtC0*vy - NtC1*vx;
    // sigma model (sxy=5.5, sdisp=0.4, baseline=1.2, focal=525)
    float sd0 = vz * (5.5f / 525.f);
    float sd2 = vz * vz * (0.4f / (525.f * 1.2f));
    float cov = (NtC0*NtC0 + NtC1*NtC1) * sd0*sd0 + NtC2*NtC2 * sd2*sd2;
    float sigma = sqrtf(cov + 1e-8f);
    float isg = 1.f / (sigma + 1e-8f);
    res *= isg;
    // JFp = [Jrot, -NtC];  J_Z = -JFp * isg
    float JZ[6] = { -Jr0*isg, -Jr1*isg, -Jr2*isg, NtC0*isg, NtC1*isg, NtC2*isg };
    res = occ ? 1e-6f : res;
    float rg = -GEOMW * res * vmask;
    float ar = fabsf(rg);
    float hw = (ar <= HUBER_D) ? 1.f : HUBER_D / fmaxf(ar, 1e-12f);
    float s  = sqrtf(hw);
    #pragma unroll
    for (int k = 0; k < 6; ++k)
      myrows[8*8 + k] = s * GEOMW * JZ[k] * vmask;
    myrows[8*8 + 6] = s * rg;
    myrows[8*8 + 7] = 0.f;
  }

  __syncthreads();

  // ---- WMMA Gram accumulation: D += A * A^T-content (symmetric) ----
  const int h = lane >> 4, idx = lane & 15;
  const int col = (idx < 7) ? idx : 7;          // lanes 7..15 read the zero pad
  const float* colptr = lds + wave * 288 * 8 + col;
  v8f d8 = { 0.f, 0.f, 0.f, 0.f, 0.f, 0.f, 0.f, 0.f };
  #pragma unroll 4
  for (int g = 0; g < 72; ++g) {
    int r0 = (g*4 + 2*h) * 8;
    v2f afrag;
    afrag[0] = colptr[r0];
    afrag[1] = colptr[r0 + 8];
    d8 = __builtin_amdgcn_wmma_f32_16x16x4_f32(
        /*neg_a=*/false, afrag, /*neg_b=*/false, afrag,
        /*c_mod=*/(short)0, d8, /*reuse_a=*/false, /*reuse_b=*/false);
  }

  // lanes 0..6 hold columns n=0..6 of rows m=VGPR 0..6 (M = vgpr, N = lane)
  if (lane < 7) {
    #pragma unroll
    for (int m = 0; m < 7; ++m)
      atomicAdd(&bacc[m*7 + lane], d8[m]);
  }
  __syncthreads();
  if (tid < 49) atomicAdd(&gacc[b*49 + tid], bacc[tid]);
}

// ---------------------------------------------------------------------------
// Zero the global accumulator
// ---------------------------------------------------------------------------
__global__ void zero_kernel(float* __restrict__ gacc) {
  int i = threadIdx.x;
  if (i < B_ * 49) gacc[i] = 0.f;
}

// ---------------------------------------------------------------------------
// Per-batch 6x6 LM solve + SE(3) exp + pose update (1 thread per batch)
// ---------------------------------------------------------------------------
__global__ void solve_kernel(const float* __restrict__ gacc,
                             float* __restrict__ poseB) {
  int b = threadIdx.x;
  if (b >= B_) return;

  float A[6][7];
  for (int i = 0; i < 6; ++i) {
    for (int j = 0; j < 6; ++j)
      A[i][j] = gacc[b*49 + i*7 + j] + ((i == j) ? LM_EPS : 0.f);
    A[i][6] = gacc[b*49 + i*7 + 6];
  }
  // Gaussian elimination with partial pivoting
  for (int k = 0; k < 6; ++k) {
    int piv = k; float mx = fabsf(A[k][k]);
    for (int i = k+1; i < 6; ++i)
      if (fabsf(A[i][k]) > mx) { mx = fabsf(A[i][k]); piv = i; }
    if (piv != k)
      for (int j = 0; j < 7; ++j) { float t = A[k][j]; A[k][j] = A[piv][j]; A[piv][j] = t; }
    float inv = 1.f / A[k][k];
    for (int i = k+1; i < 6; ++i) {
      float f = A[i][k] * inv;
      for (int j = k; j < 7; ++j) A[i][j] -= f * A[k][j];
    }
  }
  float xi[6];
  for (int i = 5; i >= 0; --i) {
    float s = A[i][6];
    for (int j = i+1; j < 6; ++j) s -= A[i][j] * xi[j];
    xi[i] = s / A[i][i];
  }

  // SE(3) exponential
  float w0 = xi[0], w1 = xi[1], w2 = xi[2];
  float u0 = xi[3], u1 = xi[4], u2 = xi[5];
  float th2 = w0*w0 + w1*w1 + w2*w2;
  bool small = th2 < 1e-8f;
  float th2s = small ? 1.f : th2;
  float th = sqrtf(th2s);
  float Af = small ? (1.f - th2/6.f)        : (sinf(th)/th);
  float Bf = small ? (0.5f - th2/24.f)      : ((1.f - cosf(th))/th2s);
  float Cf = small ? (1.f/6.f - th2/120.f)  : ((th - sinf(th))/(th2s*th));
  float K[3][3] = { {0.f, -w2,  w1}, { w2, 0.f, -w0}, {-w1,  w0, 0.f} };
  float K2[3][3];
  for (int i = 0; i < 3; ++i)
    for (int j = 0; j < 3; ++j)
      K2[i][j] = K[i][0]*K[0][j] + K[i][1]*K[1][j] + K[i][2]*K[2][j];
  float Rm[3][3], Vm[3][3];
  for (int i = 0; i < 3; ++i)
    for (int j = 0; j < 3; ++j) {
      float id = (i == j) ? 1.f : 0.f;
      Rm[i][j] = id + Af*K[i][j] + Bf*K2[i][j];
      Vm[i][j] = id + Bf*K[i][j] + Cf*K2[i][j];
    }
  float tv[3];
  for (int i = 0; i < 3; ++i)
    tv[i] = Vm[i][0]*u0 + Vm[i][1]*u1 + Vm[i][2]*u2;

  float T[4][4] = { {Rm[0][0],Rm[0][1],Rm[0][2],tv[0]},
                    {Rm[1][0],Rm[1][1],Rm[1][2],tv[1]},
                    {Rm[2][0],Rm[2][1],Rm[2][2],tv[2]},
                    {0.f, 0.f, 0.f, 1.f} };
  float Pm[4][4];
  for (int i = 0; i < 4; ++i)
    for (int j = 0; j < 4; ++j)
      Pm[i][j] = poseB[b*16 + i*4 + j];
  for (int i = 0; i < 4; ++i)
    for (int j = 0; j < 4; ++j) {
      float s = 0.f;
      for (int k = 0; k < 4; ++k) s += T[i][k] * Pm[k][j];
      poseB[b*16 + i*4 + j] = s;
    }
}

// ---------------------------------------------------------------------------
extern "C" void kernel_launch(void* const* d_in, const int* in_sizes, int n_in,
                              void* d_out, int out_size, void* d_ws, size_t ws_size,
                              hipStream_t stream) {
  (void)in_sizes; (void)n_in; (void)out_size; (void)ws_size;
  const float* pose_in = (const float*)d_in[0];
  const float* I0      = (const float*)d_in[1];
  const float* I1      = (const float*)d_in[2];
  const float* intr    = (const float*)d_in[5];
  const float* depth0  = (const float*)d_in[6];
  const float* depth1  = (const float*)d_in[7];

  float* ws = (float*)d_ws;
  size_t o = 0;
  float* V0p = ws + o; o += (size_t)3 * B_ * N_;
  float* V1p = ws + o; o += (size_t)3 * B_ * N_;
  float* N0p = ws + o; o += (size_t)3 * B_ * N_;
  float* N1p = ws + o; o += (size_t)3 * B_ * N_;
  float* GXp = ws + o; o += (size_t)C_ * B_ * N_;
  float* GYp = ws + o; o += (size_t)C_ * B_ * N_;
  float* poseW = ws + o; o += (size_t)B_ * 16;
  float* accW  = ws + o; o += (size_t)B_ * 49;

  hipMemcpyAsync(poseW, pose_in, B_ * 16 * sizeof(float),
                 hipMemcpyDeviceToDevice, stream);

  precompute_kernel<<<(B_ * N_ + 255) / 256, 256, 0, stream>>>(
      I0, depth0, depth1, intr, V0p, V1p, N0p, N1p, GXp, GYp);

  for (int it = 0; it < 3; ++it) {
    zero_kernel<<<1, 256, 0, stream>>>(accW);
    accum_kernel<<<dim3(N_ / 128, B_), 128, 0, stream>>>(
        I0, I1, depth0, depth1, intr, poseW,
        V0p, V1p, N0p, N1p, GXp, GYp, accW);
    solve_kernel<<<1, B_, 0, stream>>>(accW, poseW);
  }

  hipMemcpyAsync(d_out, poseW, B_ * 16 * sizeof(float),
                 hipMemcpyDeviceToDevice, stream);
}